// Actor_43800076484744
// MI455X (gfx1250) — compile-verified
//
#include <hip/hip_runtime.h>
#include <hip/hip_bf16.h>

typedef __attribute__((ext_vector_type(2))) float v2f;
typedef __attribute__((ext_vector_type(8))) float v8f;

#define NN 2048
#define DD 64
#define KP 4

static __device__ __forceinline__ float fexp(float x) {
    return __builtin_amdgcn_exp2f(x * 1.44269504088896340736f);
}
static __device__ __forceinline__ float frcp(float x) {
    return __builtin_amdgcn_rcpf(x);
}
// tanh(u) for u > 0: 1 - 2/(exp(2u)+1); exp overflow -> inf -> rcp -> 0 -> tanh = 1 (graceful)
static __device__ __forceinline__ float tanh_pos(float u) {
    return 1.0f - 2.0f * frcp(fexp(2.0f * u) + 1.0f);
}

// ---------------------------------------------------------------------------
// Kernel 1: AggF[N][D] = edges[N][N] @ attributes[N][D]   (fp32 WMMA 16x16x4)
// One wave per 16x16 output tile. tiles: (2048/16) x (64/16) = 128 x 4 = 512.
// ---------------------------------------------------------------------------
__global__ void k_aggf(const float* __restrict__ edges,
                       const float* __restrict__ attr,
                       float* __restrict__ aggf) {
    int wave = (blockIdx.x * blockDim.x + threadIdx.x) >> 5;
    int lane = threadIdx.x & 31;
    int tm = wave >> 2;          // 0..127
    int tn = wave & 3;           // 0..3
    int l15  = lane & 15;
    int half = lane >> 4;
    int kk   = half << 1;        // 0 or 2  (32-bit A layout: half-waves hold K pairs)
    int m    = tm * 16 + l15;    // A row for this lane
    int col  = tn * 16 + l15;    // B/C column for this lane

    v8f c = {};
    for (int k = 0; k < NN; k += 4) {
        v2f a = *(const v2f*)(edges + (size_t)m * NN + k + kk);   // b64 load
        v2f b;
        b.x = attr[(size_t)(k + kk) * DD + col];
        b.y = attr[(size_t)(k + kk + 1) * DD + col];
        c = __builtin_amdgcn_wmma_f32_16x16x4_f32(false, a, false, b,
                                                  (short)0, c, false, false);
    }
    // C layout: VGPR r -> M = r + 8*half, N = l15
    int rbase = tm * 16 + (half << 3);
#pragma unroll
    for (int r = 0; r < 8; ++r)
        aggf[(size_t)(rbase + r) * DD + col] = c[r];
}

// ---------------------------------------------------------------------------
// Kernel 2: per-row feature transform. One wave per row (lane owns cols l, l+32).
// Produces Fn[i][N][D] (row-L2-normalized next_feat) and attr_prob[N][D].
// ---------------------------------------------------------------------------
__global__ void k_feat(const float* __restrict__ attr,
                       const float* __restrict__ aggf,
                       const float* __restrict__ persona,
                       const float* __restrict__ rp,
                       const float* __restrict__ Wp,
                       const int*   __restrict__ times,
                       float* __restrict__ Fn,
                       float* __restrict__ attr_prob) {
    int wave = (blockIdx.x * blockDim.x + threadIdx.x) >> 5;
    int lane = threadIdx.x & 31;
    int row = wave;
    if (row >= NN) return;
    int t = times[0];

    float a0 = attr[(size_t)row * DD + lane];
    float a1 = attr[(size_t)row * DD + lane + 32];
    float g0 = aggf[(size_t)row * DD + lane];
    float g1 = aggf[(size_t)row * DD + lane + 32];

    float ap0 = 0.0f, ap1 = 0.0f;
#pragma unroll
    for (int i = 0; i < KP; ++i) {
        float ri = rp[i];
        float sc = Wp[i] * (1.0f - ri);
        float n0 = ri * a0 + sc * g0;
        float n1 = ri * a1 + sc * g1;
        float ss = n0 * n0 + n1 * n1;
#pragma unroll
        for (int m = 16; m >= 1; m >>= 1) ss += __shfl_xor(ss, m, 32);
        float inv = __builtin_amdgcn_rsqf(ss);
        float* fn = Fn + ((size_t)i * NN + row) * DD;
        fn[lane]      = n0 * inv;
        fn[lane + 32] = n1 * inv;
        float pti = persona[((size_t)t * NN + row) * KP + i];
        ap0 += pti * frcp(1.0f + fexp(-n0));   // sigmoid
        ap1 += pti * frcp(1.0f + fexp(-n1));
    }
    attr_prob[(size_t)row * DD + lane]      = ap0;
    attr_prob[(size_t)row * DD + lane + 32] = ap1;
}

// ---------------------------------------------------------------------------
// Kernel 3: per 16x16 tile of S_i = Fn_i @ Fn_i^T (WMMA, K=64 -> 16 wmma/i),
// fused masked tanh(exp(...)) accumulation into edges_prob. One wave per tile.
// ---------------------------------------------------------------------------
__global__ void k_gram(const float* __restrict__ Fn,
                       const float* __restrict__ edges,
                       const float* __restrict__ two_hop,
                       const float* __restrict__ persona,
                       const float* __restrict__ Tp,
                       const float* __restrict__ ep,
                       const int*   __restrict__ times,
                       float* __restrict__ edges_prob) {
    int wave = (blockIdx.x * blockDim.x + threadIdx.x) >> 5;
    int lane = threadIdx.x & 31;
    int tm = wave >> 7;            // 0..127
    int tn = wave & 127;           // 0..127
    int t = times[0];
    int l15  = lane & 15;
    int half = lane >> 4;
    int kk   = half << 1;
    int arow = tm * 16 + l15;      // Fn row feeding A
    int brow = tn * 16 + l15;      // Fn row feeding B (== column of S)
    int rbase = tm * 16 + (half << 3);
    int col   = tn * 16 + l15;

    // Hoist masks (structure-only use of adjacency) out of the i-loop.
    unsigned emask = 0, umask = 0;
#pragma unroll
    for (int r = 0; r < 8; ++r) {
        int row = rbase + r;
        float ev = edges[(size_t)row * NN + col];
        float tv = two_hop[(size_t)row * NN + col];
        if (ev > 0.0f)      emask |= 1u << r;
        if (ev + tv > 0.0f) umask |= 1u << r;   // (edges + two_hop) > 0
    }
    float p4[KP];
#pragma unroll
    for (int i = 0; i < KP; ++i)
        p4[i] = persona[((size_t)t * NN + col) * KP + i];

    float acc[8] = {};
#pragma unroll
    for (int i = 0; i < KP; ++i) {
        const float* fn = Fn + (size_t)i * NN * DD;
        v8f c = {};
#pragma unroll
        for (int k = 0; k < DD; k += 4) {
            v2f a = *(const v2f*)(fn + (size_t)arow * DD + k + kk);  // b64
            v2f b = *(const v2f*)(fn + (size_t)brow * DD + k + kk);  // b64
            c = __builtin_amdgcn_wmma_f32_16x16x4_f32(false, a, false, b,
                                                      (short)0, c, false, false);
        }
        float invT = frcp(Tp[i]);
        float ei   = ep[i];
        float pi   = p4[i];
#pragma unroll
        for (int r = 0; r < 8; ++r) {
            int row = rbase + r;
            float s = c[r];
            float val = 0.0f;
            if (row != col) {
                if (emask & (1u << r)) {
                    // one-hop: one_conn * delete
                    float u1 = fexp(s * invT) * ei;
                    float u2 = fexp((1.0f - s) * invT) * ei;
                    val = tanh_pos(u1) * tanh_pos(u2);
                } else if (umask & (1u << r)) {
                    // two-hop-only: conn
                    val = tanh_pos(fexp(s * invT) * ei);
                }
            }
            acc[r] += val * pi;
        }
    }
#pragma unroll
    for (int r = 0; r < 8; ++r)
        edges_prob[(size_t)(rbase + r) * NN + col] = acc[r];
}

// ---------------------------------------------------------------------------
extern "C" void kernel_launch(void* const* d_in, const int* in_sizes, int n_in,
                              void* d_out, int out_size, void* d_ws, size_t ws_size,
                              hipStream_t stream) {
    (void)in_sizes; (void)n_in; (void)out_size; (void)ws_size;
    const float* attributes = (const float*)d_in[0];
    const float* edges      = (const float*)d_in[1];
    const float* two_hop    = (const float*)d_in[2];
    const float* persona    = (const float*)d_in[3];
    const float* T          = (const float*)d_in[4];
    const float* e          = (const float*)d_in[5];
    const float* r          = (const float*)d_in[6];
    const float* W          = (const float*)d_in[7];
    const int*   times      = (const int*)d_in[8];

    float* edges_prob = (float*)d_out;                       // [N*N]
    float* attr_prob  = edges_prob + (size_t)NN * NN;        // [N*D]

    float* aggf = (float*)d_ws;                              // N*D floats
    float* Fn   = aggf + (size_t)NN * DD;                    // 4*N*D floats

    // 512 tile-waves -> 64 blocks of 256 (8 waves/block)
    k_aggf<<<64, 256, 0, stream>>>(edges, attributes, aggf);
    // 2048 row-waves -> 256 blocks
    k_feat<<<256, 256, 0, stream>>>(attributes, aggf, persona, r, W, times,
                                    Fn, attr_prob);
    // 128*128 = 16384 tile-waves -> 2048 blocks
    k_gram<<<2048, 256, 0, stream>>>(Fn, edges, two_hop, persona, T, e, times,
                                     edges_prob);
}